// SeqS4_72576357368288
// MI455X (gfx1250) — compile-verified
//
#include <hip/hip_runtime.h>
#include <hip/hip_bf16.h>

typedef __attribute__((ext_vector_type(2))) float v2f;
typedef __attribute__((ext_vector_type(8))) float v8f;

#define Hdim 128
#define Ndim 64
#define Ldim 4096
#define Bdim 32
#define SCOLS 48     // 32 batches + 1 ones-column + 15 pad (3 WMMA N-tiles)
#define KSPLIT 16
#define KCHUNK (Ldim / KSPLIT)   // 256

// ---------------------------------------------------------------------------
// Kernel 1: K[h,l] = 2 * Re( sum_n Bt[h,n] * exp(dtA[h,n] * l) )
// grid = (L/256, H), block = 256
// ---------------------------------------------------------------------------
__global__ void s4_build_K(const float* __restrict__ log_dt,
                           const float* __restrict__ log_A_real,
                           const float* __restrict__ A_imag,
                           const float* __restrict__ C_re,
                           const float* __restrict__ C_im,
                           float* __restrict__ Kmat) {
    __shared__ float s_dre[Ndim], s_dim[Ndim], s_bre[Ndim], s_bim[Ndim];
    const int h = blockIdx.y;
    const int l = blockIdx.x * blockDim.x + threadIdx.x;

    if (threadIdx.x < Ndim) {
        const int n = threadIdx.x;
        const float dt  = __expf(log_dt[h]);
        const float Are = -__expf(log_A_real[h * Ndim + n]);
        const float Aim = A_imag[h * Ndim + n];
        const float dre = dt * Are;
        const float dim = dt * Aim;
        // r = exp(dtA);  num = r - 1
        const float er = __expf(dre);
        float sn, cs;
        __sincosf(dim, &sn, &cs);
        const float num_re = er * cs - 1.0f;
        const float num_im = er * sn;
        // Bt = C * num / A   (complex)
        const float cre = C_re[h * Ndim + n];
        const float cim = C_im[h * Ndim + n];
        const float t_re = cre * num_re - cim * num_im;
        const float t_im = cre * num_im + cim * num_re;
        const float den = Are * Are + Aim * Aim;
        s_bre[n] = (t_re * Are + t_im * Aim) / den;
        s_bim[n] = (t_im * Are - t_re * Aim) / den;
        s_dre[n] = dre;
        s_dim[n] = dim;
    }
    __syncthreads();

    const float lf = (float)l;
    float acc = 0.0f;
#pragma unroll 4
    for (int n = 0; n < Ndim; ++n) {
        const float e = __expf(s_dre[n] * lf);
        float sn, cs;
        __sincosf(s_dim[n] * lf, &sn, &cs);
        acc = fmaf(s_bre[n],  e * cs, acc);
        acc = fmaf(-s_bim[n], e * sn, acc);
    }
    Kmat[h * Ldim + l] = 2.0f * acc;
}

// ---------------------------------------------------------------------------
// Kernel 1b: materialize Brev[l, n] (L x SCOLS, row-major):
//   n < 32  -> data[n, L-1-l]
//   n == 32 -> 1.0   (ones column: yields sum_l K[h,l])
//   else    -> 0.0
// One thread per element; keeps the WMMA loop branch-free.
// ---------------------------------------------------------------------------
__global__ void s4_build_Brev(const float* __restrict__ data,
                              float* __restrict__ Brev) {
    const int idx = blockIdx.x * blockDim.x + threadIdx.x;
    if (idx >= Ldim * SCOLS) return;
    const int n = idx % SCOLS;
    const int l = idx / SCOLS;
    float v;
    if (n < Bdim)       v = data[n * Ldim + (Ldim - 1 - l)];
    else if (n == Bdim) v = 1.0f;
    else                v = 0.0f;
    Brev[idx] = v;
}

// ---------------------------------------------------------------------------
// Kernel 1c: zero S (workspace is poisoned; GEMM accumulates atomically).
// ---------------------------------------------------------------------------
__global__ void s4_zero_S(float* __restrict__ S) {
    const int idx = blockIdx.x * blockDim.x + threadIdx.x;
    if (idx < Hdim * SCOLS) S[idx] = 0.0f;
}

// ---------------------------------------------------------------------------
// Kernel 2: S[h, n] += sum_{k in chunk} K[h,k] * Brev[k, n]
// using V_WMMA_F32_16X16X4_F32.
// grid = (H/16 = 8, SCOLS/16 = 3, KSPLIT = 16), block = 32 (one wave / tile).
// A layout (16x4 f32, ISA 7.12.2):  m = lane%16, k = vgpr + 2*(lane/16)
// B layout (4x16 f32, mirrored):    n = lane%16, k = vgpr + 2*(lane/16)
// C/D layout (16x16 f32):           m = vgpr + 8*(lane/16), n = lane%16
// K-split partials combined with global_atomic_add_f32.
// ---------------------------------------------------------------------------
__global__ void s4_gemm_wmma(const float* __restrict__ Kmat,
                             const float* __restrict__ Brev,
                             float* __restrict__ S) {
    const int lane = threadIdx.x;
    const int g = lane >> 4;        // lane-half selects k pair
    const int m = lane & 15;
    const int hbase = blockIdx.x * 16;
    const int nbase = blockIdx.y * 16;
    const int kbase = blockIdx.z * KCHUNK;

    const float* Arow = Kmat + (hbase + m) * Ldim + kbase + 2 * g;
    const float* Bcol = Brev + nbase + m + (size_t)(kbase + 2 * g) * SCOLS;

    v8f acc = {};
#pragma unroll 4
    for (int kk = 0; kk < KCHUNK; kk += 4) {
        // A: two consecutive K values -> one aligned 64-bit load
        v2f a = *(const v2f*)(Arow + kk);
        v2f b;
        b[0] = Bcol[(size_t)kk * SCOLS];
        b[1] = Bcol[(size_t)(kk + 1) * SCOLS];
        acc = __builtin_amdgcn_wmma_f32_16x16x4_f32(
            /*neg_a=*/false, a, /*neg_b=*/false, b,
            /*c_mod=*/(short)0, acc, /*reuse_a=*/false, /*reuse_b=*/false);
    }

#pragma unroll
    for (int i = 0; i < 8; ++i) {
        const int row = hbase + i + 8 * g;
        atomicAdd(&S[row * SCOLS + nbase + m], acc[i]);
    }
}

// ---------------------------------------------------------------------------
// Kernel 3: per-batch tail: skip + GELU(tanh) + GLU + output projection.
// grid = 32 (one block per batch), block = 256
// ---------------------------------------------------------------------------
__global__ void s4_tail(const float* __restrict__ S,
                        const float* __restrict__ data,
                        const float* __restrict__ W_in,
                        const float* __restrict__ b_in,
                        const float* __restrict__ Dvec,
                        const float* __restrict__ W_glu,
                        const float* __restrict__ b_glu,
                        const float* __restrict__ W_out,
                        const float* __restrict__ b_out,
                        float* __restrict__ out) {
    __shared__ float gbuf[Hdim];
    __shared__ float zbuf[2 * Hdim];
    __shared__ float red[256];

    const int b = blockIdx.x;
    const int tid = threadIdx.x;

    if (tid < Hdim) {
        const int h = tid;
        const float sval = S[h * SCOLS + b];       // sum_l K*data_rev
        const float ksum = S[h * SCOLS + Bdim];    // sum_l K (ones column)
        const float win = W_in[h];
        const float bin = b_in[h];
        const float conv = win * sval + bin * ksum;
        const float u_last = data[b * Ldim + (Ldim - 1)] * win + bin;
        const float y = conv + u_last * Dvec[h];
        // JAX default gelu (approximate=True, tanh form)
        const float t = tanhf(0.7978845608028654f * (y + 0.044715f * y * y * y));
        gbuf[h] = 0.5f * y * (1.0f + t);
    }
    __syncthreads();

    {   // z[tid] = b_glu[tid] + g . W_glu[:, tid]
        float acc = b_glu[tid];
#pragma unroll 8
        for (int h = 0; h < Hdim; ++h)
            acc = fmaf(gbuf[h], W_glu[h * (2 * Hdim) + tid], acc);
        zbuf[tid] = acc;
    }
    __syncthreads();

    float val = 0.0f;
    if (tid < Hdim) {
        const float zl = zbuf[tid];
        const float zg = zbuf[tid + Hdim];
        const float sig = 1.0f / (1.0f + __expf(-zg));
        val = zl * sig * W_out[tid];
    }
    red[tid] = val;
    __syncthreads();
    for (int s = 128; s > 0; s >>= 1) {
        if (tid < s) red[tid] += red[tid + s];
        __syncthreads();
    }
    if (tid == 0) out[b] = red[0] + b_out[0];
}

// ---------------------------------------------------------------------------
extern "C" void kernel_launch(void* const* d_in, const int* in_sizes, int n_in,
                              void* d_out, int out_size, void* d_ws, size_t ws_size,
                              hipStream_t stream) {
    const float* data       = (const float*)d_in[0];
    const float* W_in       = (const float*)d_in[1];
    const float* b_in       = (const float*)d_in[2];
    const float* log_dt     = (const float*)d_in[3];
    const float* log_A_real = (const float*)d_in[4];
    const float* A_imag     = (const float*)d_in[5];
    const float* C_re       = (const float*)d_in[6];
    const float* C_im       = (const float*)d_in[7];
    const float* Dvec       = (const float*)d_in[8];
    const float* W_glu      = (const float*)d_in[9];
    const float* b_glu      = (const float*)d_in[10];
    const float* W_out      = (const float*)d_in[11];
    const float* b_out      = (const float*)d_in[12];
    float* out = (float*)d_out;

    float* Kmat = (float*)d_ws;                    // 128*4096 f32 = 2 MB
    float* Brev = Kmat + Hdim * Ldim;              // 4096*48 f32 = 768 KB
    float* S    = Brev + Ldim * SCOLS;             // 128*48 f32

    s4_build_K<<<dim3(Ldim / 256, Hdim), 256, 0, stream>>>(
        log_dt, log_A_real, A_imag, C_re, C_im, Kmat);

    s4_build_Brev<<<(Ldim * SCOLS + 255) / 256, 256, 0, stream>>>(data, Brev);

    s4_zero_S<<<(Hdim * SCOLS + 255) / 256, 256, 0, stream>>>(S);

    s4_gemm_wmma<<<dim3(Hdim / 16, SCOLS / 16, KSPLIT), 32, 0, stream>>>(
        Kmat, Brev, S);

    s4_tail<<<Bdim, 256, 0, stream>>>(S, data, W_in, b_in, Dvec,
                                      W_glu, b_glu, W_out, b_out, out);
}